// WeightedGraphSAGE_51384988729581
// MI455X (gfx1250) — compile-verified
//
#include <hip/hip_runtime.h>
#include <hip/hip_bf16.h>

typedef __attribute__((ext_vector_type(2))) float v2f;
typedef __attribute__((ext_vector_type(8))) float v8f;

#define DIM 64
#define K2  128    // concat width
#define WPAD 130   // LDS row stride (even -> 8B-aligned b64 loads, conflict-free)

// ---------------------------------------------------------------------------
// Zero a float buffer
// ---------------------------------------------------------------------------
__global__ void gsage_zero_kernel(float* __restrict__ p, int n) {
    int i = blockIdx.x * blockDim.x + threadIdx.x;
    if (i < n) p[i] = 0.0f;
}

// ---------------------------------------------------------------------------
// Scatter: for each edge e: agg[src[e]] += x[dst[e]] * w[e]; cnt[src[e]] += 1
// 16 threads per edge, float4 per thread (coalesced gather, L2-resident).
// ---------------------------------------------------------------------------
__global__ void gsage_scatter_kernel(const float* __restrict__ xin,
                                     const int* __restrict__ src,
                                     const int* __restrict__ dst,
                                     const float* __restrict__ ew,
                                     float* __restrict__ agg,
                                     float* __restrict__ cnt,
                                     int nEdges) {
    unsigned int gid = blockIdx.x * blockDim.x + threadIdx.x;
    int e = gid >> 4;
    int q = gid & 15;
    if (e >= nEdges) return;
    int s = src[e];
    int t = dst[e];
    float w = ew[e];
    const float4 v = reinterpret_cast<const float4*>(xin + (size_t)t * DIM)[q];
    float* ap = agg + (size_t)s * DIM + q * 4;
    atomicAdd(ap + 0, v.x * w);
    atomicAdd(ap + 1, v.y * w);
    atomicAdd(ap + 2, v.z * w);
    atomicAdd(ap + 3, v.w * w);
    if (q == 0) atomicAdd(cnt + s, 1.0f);
}

// ---------------------------------------------------------------------------
// Fused concat([h, agg/cnt]) @ W^T + b, ReLU — via V_WMMA_F32_16X16X4_F32.
// One wave per 16-row node tile; 4 column tiles (N=64); K=128 in steps of 4.
// K-loop split into h-half and agg-half: no per-iteration branching.
// ---------------------------------------------------------------------------
__global__ void __launch_bounds__(256)
gsage_gemm_kernel(const float* __restrict__ hin,
                  const float* __restrict__ agg,
                  const float* __restrict__ cnt,
                  const float* __restrict__ W,    // [64,128] row-major
                  const float* __restrict__ bias, // [64]
                  float* __restrict__ out,        // [N,64]
                  int nNodes, int nTiles) {
    __shared__ float Wl[DIM * WPAD];
    __shared__ float bl[DIM];

    for (int i = threadIdx.x; i < DIM * K2; i += blockDim.x) {
        int r = i / K2, c = i - r * K2;
        Wl[r * WPAD + c] = W[i];
    }
    if (threadIdx.x < DIM) bl[threadIdx.x] = bias[threadIdx.x];
    __syncthreads();

    const int wave = threadIdx.x >> 5;
    const int lane = threadIdx.x & 31;
    const int tile = blockIdx.x * 8 + wave;
    if (tile >= nTiles) return;  // wave-uniform: EXEC stays all-ones

    const int mrow  = tile * 16 + (lane & 15);            // A-matrix row
    const int mrowc = mrow < nNodes ? mrow : (nNodes - 1);
    const int khalf = (lane >> 4) * 2;                    // K sub-offset {0,2}
    const int ncol  = lane & 15;                          // B-matrix column

    const float inv = 1.0f / fmaxf(cnt[mrowc], 1.0f);
    const v2f invv = {inv, inv};
    const float* __restrict__ hrow = hin + (size_t)mrowc * DIM + khalf;
    const float* __restrict__ arow = agg + (size_t)mrowc * DIM + khalf;
    const float* __restrict__ w0 = Wl + (ncol     ) * WPAD + khalf;
    const float* __restrict__ w1 = Wl + (ncol + 16) * WPAD + khalf;
    const float* __restrict__ w2 = Wl + (ncol + 32) * WPAD + khalf;
    const float* __restrict__ w3 = Wl + (ncol + 48) * WPAD + khalf;

    v8f acc0 = {}, acc1 = {}, acc2 = {}, acc3 = {};

    // ---- K half 1: self features h[m, 0..63] ----
    #pragma unroll 4
    for (int kk = 0; kk < DIM; kk += 4) {
        v2f a  = *(const v2f*)(hrow + kk);
        v2f b0 = *(const v2f*)(w0 + kk);
        v2f b1 = *(const v2f*)(w1 + kk);
        v2f b2 = *(const v2f*)(w2 + kk);
        v2f b3 = *(const v2f*)(w3 + kk);
        acc0 = __builtin_amdgcn_wmma_f32_16x16x4_f32(false, a, false, b0, (short)0, acc0, false, false);
        acc1 = __builtin_amdgcn_wmma_f32_16x16x4_f32(false, a, false, b1, (short)0, acc1, false, false);
        acc2 = __builtin_amdgcn_wmma_f32_16x16x4_f32(false, a, false, b2, (short)0, acc2, false, false);
        acc3 = __builtin_amdgcn_wmma_f32_16x16x4_f32(false, a, false, b3, (short)0, acc3, false, false);
    }

    // ---- K half 2: neighbor mean agg[m, 0..63] * inv ----
    #pragma unroll 4
    for (int kk = 0; kk < DIM; kk += 4) {
        v2f a  = (*(const v2f*)(arow + kk)) * invv;
        v2f b0 = *(const v2f*)(w0 + DIM + kk);
        v2f b1 = *(const v2f*)(w1 + DIM + kk);
        v2f b2 = *(const v2f*)(w2 + DIM + kk);
        v2f b3 = *(const v2f*)(w3 + DIM + kk);
        acc0 = __builtin_amdgcn_wmma_f32_16x16x4_f32(false, a, false, b0, (short)0, acc0, false, false);
        acc1 = __builtin_amdgcn_wmma_f32_16x16x4_f32(false, a, false, b1, (short)0, acc1, false, false);
        acc2 = __builtin_amdgcn_wmma_f32_16x16x4_f32(false, a, false, b2, (short)0, acc2, false, false);
        acc3 = __builtin_amdgcn_wmma_f32_16x16x4_f32(false, a, false, b3, (short)0, acc3, false, false);
    }

    // C/D layout: VGPR r -> row M = r + 8*(lane/16), col N = lane%16 (+16*nt)
    const int mbase = tile * 16 + (lane >> 4) * 8;
    #pragma unroll
    for (int r = 0; r < 8; ++r) {
        const int m = mbase + r;
        if (m < nNodes) {
            float* orow = out + (size_t)m * DIM;
            float v0 = acc0[r] + bl[ncol];      orow[ncol]      = fmaxf(v0, 0.0f);
            float v1 = acc1[r] + bl[ncol + 16]; orow[ncol + 16] = fmaxf(v1, 0.0f);
            float v2 = acc2[r] + bl[ncol + 32]; orow[ncol + 32] = fmaxf(v2, 0.0f);
            float v3 = acc3[r] + bl[ncol + 48]; orow[ncol + 48] = fmaxf(v3, 0.0f);
        }
    }
}

// ---------------------------------------------------------------------------
// Launch
// ---------------------------------------------------------------------------
extern "C" void kernel_launch(void* const* d_in, const int* in_sizes, int n_in,
                              void* d_out, int out_size, void* d_ws, size_t ws_size,
                              hipStream_t stream) {
    const float* x   = (const float*)d_in[0];
    const int*   ei  = (const int*)d_in[1];   // [2,E] row-major: first E = src
    const float* ew  = (const float*)d_in[2];
    const float* W1  = (const float*)d_in[3];
    const float* b1  = (const float*)d_in[4];
    const float* W2  = (const float*)d_in[5];
    const float* b2  = (const float*)d_in[6];
    float* out = (float*)d_out;

    const int nNodes = in_sizes[0] / DIM;
    const int nEdges = in_sizes[2];
    const int* src = ei;
    const int* dst = ei + nEdges;

    float* agg = (float*)d_ws;                 // nNodes*64
    float* cnt = agg + (size_t)nNodes * DIM;   // nNodes
    float* h1  = cnt + nNodes;                 // nNodes*64

    const int nTiles   = (nNodes + 15) / 16;
    const int zeroN    = nNodes * (DIM + 1);
    const int zeroBlk  = (zeroN + 255) / 256;
    const int scatBlk  = (nEdges * 16 + 255) / 256;
    const int gemmBlk  = (nTiles + 7) / 8;

    // ---- Layer 1 ----
    gsage_zero_kernel<<<zeroBlk, 256, 0, stream>>>(agg, zeroN);
    gsage_scatter_kernel<<<scatBlk, 256, 0, stream>>>(x, src, dst, ew, agg, cnt, nEdges);
    gsage_gemm_kernel<<<gemmBlk, 256, 0, stream>>>(x, agg, cnt, W1, b1, h1, nNodes, nTiles);

    // ---- Layer 2 ----
    gsage_zero_kernel<<<zeroBlk, 256, 0, stream>>>(agg, zeroN);
    gsage_scatter_kernel<<<scatBlk, 256, 0, stream>>>(h1, src, dst, ew, agg, cnt, nEdges);
    gsage_gemm_kernel<<<gemmBlk, 256, 0, stream>>>(h1, agg, cnt, W2, b2, out, nNodes, nTiles);
}